// GCN_84524956385672
// MI455X (gfx1250) — compile-verified
//
#include <hip/hip_runtime.h>
#include <hip/hip_bf16.h>

#define N_NODES 100000
#define N_EDGES 1600000
#define FEAT    32
#define EMBD    32
#define HID     64
#define NGRAPH  128

typedef float v2f __attribute__((ext_vector_type(2)));
typedef float v8f __attribute__((ext_vector_type(8)));

static __device__ __forceinline__ void atomAddF(float* p, float v) {
    (void)__hip_atomic_fetch_add(p, v, __ATOMIC_RELAXED, __HIP_MEMORY_SCOPE_AGENT);
}

// ---------------- h0 = concat(x, emb[mapping]) ----------------
__global__ void k_concat(const float* __restrict__ x, const int* __restrict__ map,
                         const float* __restrict__ emb, float* __restrict__ h) {
    int i = blockIdx.x * blockDim.x + threadIdx.x;
    if (i >= N_NODES * HID) return;
    int node = i >> 6, c = i & 63;
    h[i] = (c < FEAT) ? x[(size_t)node * FEAT + c]
                      : emb[(size_t)map[node] * EMBD + (c - FEAT)];
}

// ---------------- degree: self-loop weight 1 ----------------
__global__ void k_deg_init(float* __restrict__ deg) {
    int i = blockIdx.x * blockDim.x + threadIdx.x;
    if (i < N_NODES) deg[i] = 1.0f;
}
__global__ void k_deg_edges(const int* __restrict__ dst, const float* __restrict__ ew,
                            float* __restrict__ deg) {
    int e = blockIdx.x * blockDim.x + threadIdx.x;
    if (e < N_EDGES) atomAddF(&deg[dst[e]], ew[e]);
}
__global__ void k_deg_finalize(float* __restrict__ deg) {
    int i = blockIdx.x * blockDim.x + threadIdx.x;
    if (i >= N_NODES) return;
    float d = deg[i];
    deg[i] = (d > 0.0f) ? rsqrtf(d) : 0.0f;   // becomes dinv in place
}

// ---------------- WMMA f32 GEMM: C[N,64] = A[N,64] x W[64,64] ----------------
// One wave = 16-row x 64-col tile. 4 waves/block -> 64 rows/block.
__global__ __launch_bounds__(128) void k_gemm_wmma(const float* __restrict__ A,
                                                   const float* __restrict__ W,
                                                   float* __restrict__ C, int n_rows) {
    const int lane  = threadIdx.x;          // 0..31
    const int wave  = threadIdx.y;          // 0..3
    const int base  = (blockIdx.x * 4 + wave) * 16;
    const int mcol  = lane & 15;            // row (A) / column (B,D) index in half-wave
    const int khalf = lane >> 4;            // 0: K pair {0,1}; 1: K pair {2,3}

    int arow = base + mcol;
    int arow_c = arow < n_rows ? arow : (n_rows - 1);   // clamp: keeps EXEC all-1s
    const float* arp = A + (size_t)arow_c * HID;

    v8f acc0 = {}, acc1 = {}, acc2 = {}, acc3 = {};

    #pragma unroll
    for (int ks = 0; ks < 16; ++ks) {
        const int k0 = ks * 4 + khalf * 2;
        v2f a;
        a.x = arp[k0];
        a.y = arp[k0 + 1];
        const float* wr0 = W + (size_t)k0 * HID;
        const float* wr1 = W + (size_t)(k0 + 1) * HID;
        v2f b0; b0.x = wr0[mcol +  0]; b0.y = wr1[mcol +  0];
        v2f b1; b1.x = wr0[mcol + 16]; b1.y = wr1[mcol + 16];
        v2f b2; b2.x = wr0[mcol + 32]; b2.y = wr1[mcol + 32];
        v2f b3; b3.x = wr0[mcol + 48]; b3.y = wr1[mcol + 48];
        acc0 = __builtin_amdgcn_wmma_f32_16x16x4_f32(false, a, false, b0, (short)0, acc0, false, false);
        acc1 = __builtin_amdgcn_wmma_f32_16x16x4_f32(false, a, false, b1, (short)0, acc1, false, false);
        acc2 = __builtin_amdgcn_wmma_f32_16x16x4_f32(false, a, false, b2, (short)0, acc2, false, false);
        acc3 = __builtin_amdgcn_wmma_f32_16x16x4_f32(false, a, false, b3, (short)0, acc3, false, false);
    }

    // D layout: VGPR r -> lanes 0-15 row base+r, lanes 16-31 row base+r+8, col = mcol + 16*t
    #pragma unroll
    for (int r = 0; r < 8; ++r) {
        int row = base + r + khalf * 8;
        if (row < n_rows) {
            float* cp = C + (size_t)row * HID + mcol;
            cp[ 0] = acc0[r];
            cp[16] = acc1[r];
            cp[32] = acc2[r];
            cp[48] = acc3[r];
        }
    }
}

// ---------------- out = bias + self-loop (dinv^2) * hw ----------------
__global__ void k_outinit(const float* __restrict__ hw, const float* __restrict__ dinv,
                          const float* __restrict__ bias, float* __restrict__ out) {
    int i = blockIdx.x * blockDim.x + threadIdx.x;
    if (i >= N_NODES * HID) return;
    int node = i >> 6, c = i & 63;
    float di = dinv[node];
    out[i] = bias[c] + di * di * hw[i];
}

// ---------------- edge scatter: out[dst] += norm_e * hw[src] ----------------
// 16 lanes per edge, 4 channels each (float4 gather + 4 hw f32 atomics)
__global__ void k_msg(const int* __restrict__ src, const int* __restrict__ dst,
                      const float* __restrict__ ew, const float* __restrict__ dinv,
                      const float* __restrict__ hw, float* __restrict__ out) {
    long long idx = (long long)blockIdx.x * blockDim.x + threadIdx.x;
    int e = (int)(idx >> 4);
    if (e >= N_EDGES) return;
    int c = (int)(idx & 15) << 2;
    int s = src[e], d = dst[e];
    float nrm = dinv[s] * ew[e] * dinv[d];
    const float4 v = *(const float4*)(hw + (size_t)s * HID + c);
    float* op = out + (size_t)d * HID + c;
    atomAddF(op + 0, nrm * v.x);
    atomAddF(op + 1, nrm * v.y);
    atomAddF(op + 2, nrm * v.z);
    atomAddF(op + 3, nrm * v.w);
}

// ---------------- ReLU in place ----------------
__global__ void k_relu(float* __restrict__ h) {
    int i = blockIdx.x * blockDim.x + threadIdx.x;
    if (i < N_NODES * HID) h[i] = fmaxf(h[i], 0.0f);
}

// ---------------- mean pool ----------------
__global__ void k_pool_zero(float* __restrict__ pool, float* __restrict__ cnt) {
    int i = blockIdx.x * blockDim.x + threadIdx.x;
    if (i < NGRAPH * HID) pool[i] = 0.0f;
    if (i < NGRAPH) cnt[i] = 0.0f;
}
__global__ void k_pool_accum(const float* __restrict__ h, const int* __restrict__ batch,
                             float* __restrict__ pool, float* __restrict__ cnt) {
    int i = blockIdx.x * blockDim.x + threadIdx.x;
    if (i >= N_NODES * HID) return;
    int node = i >> 6, c = i & 63;
    int g = batch[node];
    atomAddF(&pool[(size_t)g * HID + c], h[i]);
    if (c == 0) atomAddF(&cnt[g], 1.0f);
}
__global__ void k_pool_final(const float* __restrict__ pool, const float* __restrict__ cnt,
                             float* __restrict__ out) {
    int i = blockIdx.x * blockDim.x + threadIdx.x;
    if (i >= NGRAPH * HID) return;
    out[i] = pool[i] / fmaxf(cnt[i >> 6], 1.0f);
}

extern "C" void kernel_launch(void* const* d_in, const int* in_sizes, int n_in,
                              void* d_out, int out_size, void* d_ws, size_t ws_size,
                              hipStream_t stream) {
    const float* x          = (const float*)d_in[0];
    const int*   mapping    = (const int*)d_in[1];
    const int*   edge_index = (const int*)d_in[2];
    const float* edge_attr  = (const float*)d_in[3];
    const int*   batch      = (const int*)d_in[4];
    const float* emb        = (const float*)d_in[5];
    const float* W0         = (const float*)d_in[6];
    const float* b0         = (const float*)d_in[7];
    const float* W1         = (const float*)d_in[8];
    const float* b1         = (const float*)d_in[9];
    float* out = (float*)d_out;

    const int* src = edge_index;
    const int* dst = edge_index + N_EDGES;

    // workspace layout (floats)
    float* ws   = (float*)d_ws;
    float* hA   = ws;                                  // N*64
    float* hB   = hA + (size_t)N_NODES * HID;          // N*64 (hw)
    float* hC   = hB + (size_t)N_NODES * HID;          // N*64
    float* dinv = hC + (size_t)N_NODES * HID;          // N   (deg then dinv)
    float* pool = dinv + N_NODES;                      // G*64
    float* cnt  = pool + (size_t)NGRAPH * HID;         // G

    const int T = 256;
    const int NC_BLK   = (N_NODES * HID + T - 1) / T;
    const int N_BLK    = (N_NODES + T - 1) / T;
    const int E_BLK    = (N_EDGES + T - 1) / T;
    const int MSG_BLK  = (int)(((long long)N_EDGES * 16 + T - 1) / T);
    const int GEMM_BLK = (N_NODES + 63) / 64;

    // h0 = concat(x, emb[mapping])
    k_concat<<<NC_BLK, T, 0, stream>>>(x, mapping, emb, hA);

    // degree / dinv
    k_deg_init<<<N_BLK, T, 0, stream>>>(dinv);
    k_deg_edges<<<E_BLK, T, 0, stream>>>(dst, edge_attr, dinv);
    k_deg_finalize<<<N_BLK, T, 0, stream>>>(dinv);

    // ---- layer 0: hA -> hC ----
    k_gemm_wmma<<<GEMM_BLK, dim3(32, 4), 0, stream>>>(hA, W0, hB, N_NODES);
    k_outinit<<<NC_BLK, T, 0, stream>>>(hB, dinv, b0, hC);
    k_msg<<<MSG_BLK, T, 0, stream>>>(src, dst, edge_attr, dinv, hB, hC);
    k_relu<<<NC_BLK, T, 0, stream>>>(hC);

    // ---- layer 1: hC -> hA ----
    k_gemm_wmma<<<GEMM_BLK, dim3(32, 4), 0, stream>>>(hC, W1, hB, N_NODES);
    k_outinit<<<NC_BLK, T, 0, stream>>>(hB, dinv, b1, hA);
    k_msg<<<MSG_BLK, T, 0, stream>>>(src, dst, edge_attr, dinv, hB, hA);
    k_relu<<<NC_BLK, T, 0, stream>>>(hA);

    // ---- global mean pool ----
    k_pool_zero<<<(NGRAPH * HID + T - 1) / T, T, 0, stream>>>(pool, cnt);
    k_pool_accum<<<NC_BLK, T, 0, stream>>>(hA, batch, pool, cnt);
    k_pool_final<<<(NGRAPH * HID + T - 1) / T, T, 0, stream>>>(pool, cnt, out);
}